// CognitiveManifold_35424890257809
// MI455X (gfx1250) — compile-verified
//
#include <hip/hip_runtime.h>

typedef __attribute__((ext_vector_type(2))) float v2f;
typedef __attribute__((ext_vector_type(4))) float v4f;
typedef __attribute__((ext_vector_type(8))) float v8f;

#define EPSV 1e-6f

// ---- CDNA5 async global->LDS helpers (gfx1250; ASYNCcnt-tracked, no VGPR staging) ----
__device__ __forceinline__ unsigned lds_off_of(const void* p) {
  // addrspace(3) -> generic cast keeps the LDS byte offset in the low 32 bits
  return (unsigned)(uintptr_t)p;
}
__device__ __forceinline__ void async_ld_b128(unsigned lds_off, const void* gsrc) {
  asm volatile("global_load_async_to_lds_b128 %0, %1, off"
               :: "v"(lds_off), "v"(gsrc)
               : "memory");
}
__device__ __forceinline__ void wait_async0() {
  asm volatile("s_wait_asynccnt 0" ::: "memory");
}

// ---------------- K1: mean-pool + MLP layers 1,2 ----------------
__global__ __launch_bounds__(256) void k1_pool_mlp(
    const float* __restrict__ x, const float* __restrict__ W1, const float* __restrict__ b1,
    const float* __restrict__ W2, const float* __restrict__ b2, float* __restrict__ ws_h2) {
  const int b = blockIdx.x, tid = threadIdx.x;
  __shared__ float hbar[64];
  __shared__ float h1[512];
  if (tid < 64) {
    float acc = 0.f;
    const float* xb = x + (size_t)b * 128 * 64 + tid;
    for (int s = 0; s < 128; ++s) acc += xb[s * 64];
    hbar[tid] = acc * (1.0f / 128.0f);
  }
  __syncthreads();
  for (int o = tid; o < 512; o += 256) {
    float acc = b1[o];
    for (int d = 0; d < 64; ++d) acc += hbar[d] * W1[d * 512 + o];
    h1[o] = fmaxf(acc, 0.f);
  }
  __syncthreads();
  for (int o = tid; o < 512; o += 256) {
    float acc = b2[o];
    for (int m = 0; m < 512; ++m) acc += h1[m] * W2[m * 512 + o];
    ws_h2[b * 512 + o] = fmaxf(acc, 0.f);
  }
}

// ---------------- K2: L = tril(h2@W3+b3), G = L L^T + eps I ----------------
__global__ __launch_bounds__(1024) void k2_L_G(
    const float* __restrict__ ws_h2, const float* __restrict__ W3,
    const float* __restrict__ b3, float* __restrict__ G_out) {
  const int b = blockIdx.x, tid = threadIdx.x;
  __shared__ float h2[512];
  __shared__ float L[4096];
  if (tid < 512) h2[tid] = ws_h2[b * 512 + tid];
  __syncthreads();
  #pragma unroll
  for (int q = 0; q < 4; ++q) {
    int e = tid + q * 1024;
    int r = e >> 6, c = e & 63;
    float val = 0.f;
    if (c <= r) {
      float acc = b3[e];
      for (int m = 0; m < 512; ++m) acc += h2[m] * W3[(size_t)m * 4096 + e];
      val = acc;
    }
    L[e] = val;
  }
  __syncthreads();
  #pragma unroll
  for (int q = 0; q < 4; ++q) {
    int g = tid + q * 1024;
    int r = g >> 6, c = g & 63;
    float acc = (r == c) ? EPSV : 0.f;
    for (int t = 0; t < 64; ++t) acc += L[r * 64 + t] * L[c * 64 + t];
    G_out[(size_t)b * 4096 + g] = acc;
  }
}

// ---------------- K3: Gauss-Jordan inverse (G is SPD -> no pivoting) ----------------
__global__ __launch_bounds__(256) void k3_inv(
    const float* __restrict__ G, float* __restrict__ Ginv, float* __restrict__ s_out) {
  const int b = blockIdx.x, tid = threadIdx.x;
  __shared__ float aug[64 * 129];
  __shared__ float fcol[64];
  const float* Gb = G + (size_t)b * 4096;
  for (int e = tid; e < 8192; e += 256) {
    int r = e >> 7, c = e & 127;
    aug[r * 129 + c] = (c < 64) ? Gb[r * 64 + c] : ((c - 64) == r ? 1.f : 0.f);
  }
  for (int k = 0; k < 64; ++k) {
    __syncthreads();
    float pivinv = 1.0f / aug[k * 129 + k];
    __syncthreads();
    if (tid < 128) {
      aug[k * 129 + tid] *= pivinv;
    } else if (tid < 192) {
      int r = tid - 128;
      fcol[r] = (r == k) ? 0.f : aug[r * 129 + k];
    }
    __syncthreads();
    for (int e = tid; e < 8192; e += 256) {
      int r = e >> 7, c = e & 127;
      if (r != k) aug[r * 129 + c] -= fcol[r] * aug[k * 129 + c];
    }
  }
  __syncthreads();
  for (int e = tid; e < 4096; e += 256) {
    int r = e >> 6, c = e & 63;
    Ginv[(size_t)b * 4096 + e] = aug[r * 129 + 64 + c];
  }
  if (tid < 64) {
    float acc = 0.f;
    for (int c = 0; c < 64; ++c) acc += aug[tid * 129 + 64 + c];
    s_out[b * 64 + tid] = acc;
  }
}

// ---------------- K4a: T1 = Ginv@roll(G,1,ax1), T2 = Ginv@roll(G,1,ax2)^T ----------------
__global__ __launch_bounds__(256) void k4a_T(
    const float* __restrict__ G, const float* __restrict__ Ginv,
    float* __restrict__ T1, float* __restrict__ T2) {
  const int b = blockIdx.x, tid = threadIdx.x;
  __shared__ float lG[4096];
  __shared__ float lGi[4096];
  for (int e = tid; e < 4096; e += 256) {
    lG[e] = G[(size_t)b * 4096 + e];
    lGi[e] = Ginv[(size_t)b * 4096 + e];
  }
  __syncthreads();
  for (int e = tid; e < 4096; e += 256) {
    int i = e >> 6, k = e & 63;  // second index = k for T1, = j for T2
    float a1 = 0.f, a2 = 0.f;
    for (int m = 0; m < 64; ++m) {
      int mm = (m + 63) & 63;            // (m-1) mod 64
      float gi = lGi[i * 64 + m];
      a1 += gi * lG[mm * 64 + k];        // A[m,k]  = G[(m-1)%d, k]
      a2 += gi * lG[k * 64 + mm];        // Bm[j,m] = G[j, (m-1)%d]
    }
    T1[(size_t)b * 4096 + e] = a1;
    T2[(size_t)b * 4096 + e] = a2;
  }
}

// ---------------- K4b: Gamma[b,i,j,k] = 0.5*(T1[i,k]+T2[i,j]-s[i]*G[j,k]) ----------------
__global__ __launch_bounds__(256) void k4b_gamma(
    const float* __restrict__ G, const float* __restrict__ T1, const float* __restrict__ T2,
    const float* __restrict__ s, float* __restrict__ Gamma) {
  const int g = blockIdx.x * 256 + threadIdx.x;  // < 1048576
  const int b = g >> 18;
  const int rem = g & 262143;
  const int i = rem >> 12, j = (rem >> 6) & 63, k = rem & 63;
  Gamma[g] = 0.5f * (T1[(size_t)b * 4096 + i * 64 + k] + T2[(size_t)b * 4096 + i * 64 + j]
                     - s[b * 64 + i] * G[(size_t)b * 4096 + j * 64 + k]);
}

// ---------------- K5: Riemann mega-kernel, one (b,i,j) tile per workgroup ----------------
// C[k,l] = sum_m Gamma[b,i,m,k] * Gamma[b,m,j,l]   (64x64x64 GEMM via v_wmma_f32_16x16x4_f32)
// riem[b,i,j,k,l] = Dp[l] - Dp[k] + C[k,l] - C[l,k]
// Operand tiles staged with CDNA5 async global->LDS DMA (ASYNCcnt).
__global__ __launch_bounds__(128) void k5_riemann(
    const float* __restrict__ Gamma, float* __restrict__ riem) {
  const int j = blockIdx.x, i = blockIdx.y, b = blockIdx.z;
  const int tid = threadIdx.x;
  __shared__ __align__(16) float lA[4096];  // Gamma[b,i,m,k]   (m-major)
  __shared__ __align__(16) float lB[4096];  // Gamma[b,m,j,l]   (m-major)
  __shared__ __align__(16) float lC[4096];  // C[k,l]
  __shared__ float ldp[64];

  const float* gb = Gamma + (size_t)b * 262144;

  // A tile: Gamma[b,i,:,:] is 16 KB contiguous -> 8 async b128 per thread
  const v4f* srcA = (const v4f*)(gb + (size_t)i * 4096);
  #pragma unroll
  for (int q = 0; q < 8; ++q) {
    int idx = tid + q * 128;
    async_ld_b128(lds_off_of(&lA[idx * 4]), (const void*)(srcA + idx));
  }
  // B tile: row m is Gamma[b,m,j,0..63] (64 contiguous floats), 64 rows
  #pragma unroll
  for (int q = 0; q < 8; ++q) {
    int v = tid + q * 128;
    int m = v >> 4, lq = v & 15;
    const v4f* srcB = (const v4f*)(gb + (size_t)m * 4096 + j * 64);
    async_ld_b128(lds_off_of(&lB[v * 4]), (const void*)(srcB + lq));
  }
  if (tid < 64) {
    int bm1 = (b + 3) & 3;  // roll(Gamma,1,axis=0)[b] = Gamma[b-1]
    ldp[tid] = Gamma[(size_t)bm1 * 262144 + (size_t)i * 4096 + j * 64 + tid]
             - gb[(size_t)i * 4096 + j * 64 + tid];
  }
  wait_async0();
  __syncthreads();

  const int wave = tid >> 5, lane = tid & 31;
  const int half = lane >> 4, lr = lane & 15;
  const int k0 = wave * 16;  // this wave owns C rows [k0, k0+16)

#if __has_builtin(__builtin_amdgcn_wmma_f32_16x16x4_f32)
  v8f z = {0.f, 0.f, 0.f, 0.f, 0.f, 0.f, 0.f, 0.f};
  v8f acc[4] = {z, z, z, z};
  #pragma unroll
  for (int st = 0; st < 16; ++st) {
    const int ma = st * 4 + 2 * half;  // lane's K pair within this 4-wide K step
    v2f a;
    a.x = lA[ma * 64 + k0 + lr];       // A_op[M=k', K=m'] = Gamma[b,i,m,k]
    a.y = lA[(ma + 1) * 64 + k0 + lr];
    #pragma unroll
    for (int lt = 0; lt < 4; ++lt) {
      v2f bv;
      bv.x = lB[ma * 64 + lt * 16 + lr];        // B_op[K=m', N=l'] = Gamma[b,m,j,l]
      bv.y = lB[(ma + 1) * 64 + lt * 16 + lr];
      acc[lt] = __builtin_amdgcn_wmma_f32_16x16x4_f32(
          false, a, false, bv, (short)0, acc[lt], false, false);
    }
  }
  // C/D layout: VGPR v -> row M = v + 8*half, col N = lr
  #pragma unroll
  for (int lt = 0; lt < 4; ++lt) {
    #pragma unroll
    for (int v = 0; v < 8; ++v) {
      lC[(k0 + v + 8 * half) * 64 + lt * 16 + lr] = acc[lt][v];
    }
  }
#else
  // Scalar fallback (should not be taken on gfx1250; histogram will show wmma=0)
  {
    const int k = tid >> 1, lb = (tid & 1) * 32;
    for (int l = lb; l < lb + 32; ++l) {
      float acc = 0.f;
      for (int m = 0; m < 64; ++m) acc += lA[m * 64 + k] * lB[m * 64 + l];
      lC[k * 64 + l] = acc;
    }
  }
#endif
  __syncthreads();

  // Epilogue: antisymmetrize + Dp terms, coalesced float4 stores
  float* outp = riem + (size_t)((b * 64 + i) * 64 + j) * 4096;
  const int k = tid >> 1;
  const int lbase = (tid & 1) * 32;
  const float dpk = ldp[k];
  #pragma unroll
  for (int q = 0; q < 8; ++q) {
    int l = lbase + q * 4;
    v4f val;
    #pragma unroll
    for (int u = 0; u < 4; ++u) {
      val[u] = ldp[l + u] - dpk + lC[k * 64 + l + u] - lC[(l + u) * 64 + k];
    }
    *((v4f*)(outp + k * 64 + l)) = val;
  }
}

// ---------------- K6: ricci[b,i,j] = sum_k riem[b,k,i,k,j] ----------------
__global__ __launch_bounds__(64) void k6_ricci(
    const float* __restrict__ riem, float* __restrict__ ricci) {
  const int i = blockIdx.x, b = blockIdx.y;
  const int jt = threadIdx.x;
  float acc = 0.f;
  for (int k = 0; k < 64; ++k) {
    acc += riem[(size_t)b * 16777216 + (size_t)k * 262144 + (size_t)i * 4096
                + (size_t)k * 64 + jt];
  }
  ricci[(size_t)(b * 64 + i) * 64 + jt] = acc;
}

// ---------------- K7: scalar_curv[b] = sum_ij ricci*Ginv ----------------
__global__ __launch_bounds__(256) void k7_scalar(
    const float* __restrict__ ricci, const float* __restrict__ Ginv, float* __restrict__ scal) {
  const int b = blockIdx.x, tid = threadIdx.x;
  __shared__ float red[256];
  float acc = 0.f;
  for (int e = tid; e < 4096; e += 256)
    acc += ricci[(size_t)b * 4096 + e] * Ginv[(size_t)b * 4096 + e];
  red[tid] = acc;
  __syncthreads();
  for (int off = 128; off > 0; off >>= 1) {
    if (tid < off) red[tid] += red[tid + off];
    __syncthreads();
  }
  if (tid == 0) scal[b] = red[0];
}

// ---------------- K8: passthrough copy of x ----------------
__global__ __launch_bounds__(256) void k8_copy(const float* __restrict__ x, float* __restrict__ o) {
  const int g = blockIdx.x * 256 + threadIdx.x;
  o[g] = x[g];
}

extern "C" void kernel_launch(void* const* d_in, const int* in_sizes, int n_in,
                              void* d_out, int out_size, void* d_ws, size_t ws_size,
                              hipStream_t stream) {
  const float* x  = (const float*)d_in[0];
  const float* W1 = (const float*)d_in[1];
  const float* b1 = (const float*)d_in[2];
  const float* W2 = (const float*)d_in[3];
  const float* b2 = (const float*)d_in[4];
  const float* W3 = (const float*)d_in[5];
  const float* b3 = (const float*)d_in[6];

  float* out = (float*)d_out;
  // Output layout (floats): G | Gamma | riemann | ricci | scalar | x
  float* G_out     = out + 0;
  float* Gamma_out = out + 16384;
  float* riem_out  = out + 1064960;    // 16384 + 1048576
  float* ricci_out = out + 68173824;   // + 4*64^4
  float* scal_out  = out + 68190208;
  float* x_out     = out + 68190212;

  float* ws = (float*)d_ws;
  float* ws_h2   = ws;            // 4*512   = 2048
  float* ws_Ginv = ws + 2048;     // 4*4096  = 16384
  float* ws_s    = ws + 18432;    // 4*64    = 256
  float* ws_T1   = ws + 18688;    // 4*4096  = 16384
  float* ws_T2   = ws + 35072;    // 4*4096  = 16384  (total ~206 KB)

  k1_pool_mlp<<<4, 256, 0, stream>>>(x, W1, b1, W2, b2, ws_h2);
  k2_L_G<<<4, 1024, 0, stream>>>(ws_h2, W3, b3, G_out);
  k3_inv<<<4, 256, 0, stream>>>(G_out, ws_Ginv, ws_s);
  k4a_T<<<4, 256, 0, stream>>>(G_out, ws_Ginv, ws_T1, ws_T2);
  k4b_gamma<<<4096, 256, 0, stream>>>(G_out, ws_T1, ws_T2, ws_s, Gamma_out);
  k5_riemann<<<dim3(64, 64, 4), 128, 0, stream>>>(Gamma_out, riem_out);
  k6_ricci<<<dim3(64, 4), 64, 0, stream>>>(riem_out, ricci_out);
  k7_scalar<<<4, 256, 0, stream>>>(ricci_out, ws_Ginv, scal_out);
  k8_copy<<<128, 256, 0, stream>>>(x, x_out);
}